// HMaxPool2D_67010079752689
// MI455X (gfx1250) — compile-verified
//
#include <hip/hip_runtime.h>
#include <hip/hip_bf16.h>

// Hex max-pool, closed form for (B,H,W,C)=(32,128,128,64) -> (32,43,43,64).
// Center for output (i,j): row=3i, col=3j+(i&1).
// i even -> MASK_ODD  (skip dx=-1 taps on dy=+/-1 rows)
// i odd  -> MASK_EVEN (skip dx=+1 taps on dy=+/-1 rows)
// Masked / zero-padded cells contribute exactly 0.0 -> acc starts at 0.
//
// Grid carries (b, i) in (z, y) so no integer division is needed and all
// i-dependent control flow is wave-uniform (scalar branches).

typedef __attribute__((ext_vector_type(4))) float v4f;

__global__ __launch_bounds__(256) void hexpool_kernel(const float* __restrict__ x,
                                                      float* __restrict__ out) {
    constexpr int H = 128, W = 128, C = 64, OH = 43, OW = 43;

    const int tid = threadIdx.x;
    const int j  = blockIdx.x * 16 + (tid >> 4);  // 16 spatial sites per block
    const int i  = blockIdx.y;                    // uniform per block
    const int b  = blockIdx.z;                    // uniform per block
    if (j >= OW) return;

    const int c0 = (tid & 15) << 2;               // 4 channels per thread (float4)

    const int baseR  = 3 * i;                     // uniform
    const int baseC  = 3 * j + (i & 1);
    const int skipDx = (i & 1) ? 1 : -1;          // uniform: masked corner column

    const v4f* __restrict__ xv = reinterpret_cast<const v4f*>(x);
    const long imgBase = (long)b * (H * W);       // row 0 of this image, in pixels

    v4f acc;
    acc.x = 0.0f; acc.y = 0.0f; acc.z = 0.0f; acc.w = 0.0f;

#pragma unroll
    for (int dy = -1; dy <= 1; ++dy) {
        const int rr = baseR + dy;                // in [-1,127]; uniform test
        if (rr < 0) continue;                     // only i==0, dy==-1 (scalar branch)
        const long rowBase = imgBase + (long)rr * W;
#pragma unroll
        for (int dx = -1; dx <= 1; ++dx) {
            if (dy != 0 && dx == skipDx) continue;     // uniform (scalar branch)
            const int cc = baseC + dx;                 // in [-1,128]
            if ((unsigned)cc >= (unsigned)W) continue; // zero padding -> contributes 0
            const v4f v = xv[(((rowBase + cc) * C) + c0) >> 2];
            acc.x = fmaxf(acc.x, v.x);
            acc.y = fmaxf(acc.y, v.y);
            acc.z = fmaxf(acc.z, v.z);
            acc.w = fmaxf(acc.w, v.w);
        }
    }

    // out[((b*OH + i)*OW + j)*C + c0 .. +3], streaming store (write-once data).
    const long o = (((long)(b * OH + i) * OW + j) * C + c0) >> 2;
    v4f* __restrict__ ov = reinterpret_cast<v4f*>(out);
    __builtin_nontemporal_store(acc, &ov[o]);
}

extern "C" void kernel_launch(void* const* d_in, const int* in_sizes, int n_in,
                              void* d_out, int out_size, void* d_ws, size_t ws_size,
                              hipStream_t stream) {
    (void)in_sizes; (void)n_in; (void)out_size; (void)d_ws; (void)ws_size;
    const float* x = (const float*)d_in[0];
    float* out = (float*)d_out;

    dim3 grid((43 + 15) / 16, 43, 32);            // (j-tiles, i, b)
    hexpool_kernel<<<grid, 256, 0, stream>>>(x, out);
}